// TI_Loss_31421980738233
// MI455X (gfx1250) — compile-verified
//
#include <hip/hip_runtime.h>

typedef float v2f __attribute__((ext_vector_type(2)));
typedef float v8f __attribute__((ext_vector_type(8)));

#define PAD_ID 0
#define UNK_ID 1
#define END_ID 2

// ---------------------------------------------------------------------------
// Kernel 1: one block per batch row. Finds first PAD, then every thread
// computes the masked NLL contribution and the 0/1 mask for its position.
// Latency-bound scattered gather from the 1GB logits tensor -> maximize
// thread-level parallelism (8192 concurrent gathers).
// ---------------------------------------------------------------------------
__global__ void ti_loss_gather(const float* __restrict__ logits,
                               const int* __restrict__ ftrgs,
                               const int* __restrict__ targets,
                               float* __restrict__ wsN,   // [B, L] masked nll
                               float* __restrict__ wsM,   // [B, L] 0/1 mask
                               int L, long long V)
{
    const int b = blockIdx.x;
    const int l = threadIdx.x;           // blockDim.x == L
    __shared__ int s_first;
    if (l == 0) s_first = L;
    __syncthreads();

    const int ft = ftrgs[b * L + l];
    if (ft == PAD_ID) atomicMin(&s_first, l);
    __syncthreads();

    const int  firstPad = s_first;
    const bool unk      = (l < firstPad) && (ft == UNK_ID);

    float val = 0.0f, msk = 0.0f;
    if (unk) {
        // prev_tgt = roll(targets,1): position 0 wraps to targets[L-1]
        const int prev = targets[b * L + (l == 0 ? (L - 1) : (l - 1))];
        const float p  = logits[(long long)(b * L + l) * V + prev];
        val = -logf(p);
        msk = 1.0f;
    }
    wsN[b * L + l] = val;
    wsM[b * L + l] = msk;
}

// ---------------------------------------------------------------------------
// Kernel 2: single block, 4 waves. Reduces both [16,L] arrays with chained
// V_WMMA_F32_16X16X4_F32 (B = all-ones => D[m,n] += sum_k A[m,k], f32-exact
// row sums, M dimension = batch). Wave w covers positions [w*L/4, (w+1)*L/4).
// A-matrix 16x4 f32 layout: lanes 0-15 hold (M=lane, K=0..1), lanes 16-31
// hold (M=lane-16, K=2..3)  => per-lane float2 at offset 4c + 2*(lane>>4).
// C/D layout: lane 0 VGPR j = D[j,0] (batches 0-7), lane 16 VGPR j = D[j+8,0].
// ---------------------------------------------------------------------------
__global__ void ti_loss_reduce_wmma(const float* __restrict__ wsN,
                                    const float* __restrict__ wsM,
                                    const float* __restrict__ logits,
                                    const int* __restrict__ seqlen,
                                    const int* __restrict__ inserted,
                                    float* __restrict__ out,
                                    int L, long long V)
{
    __shared__ float sN[16];     // per-batch sum of masked nll
    __shared__ float sC[16];     // per-batch unk count (float, exact)
    __shared__ float sLoss[16];
    __shared__ float sAct[16];

    const int tid  = threadIdx.x;        // 128 threads = 4 waves
    const int lane = tid & 31;
    const int wave = tid >> 5;

    if (tid < 16) { sN[tid] = 0.0f; sC[tid] = 0.0f; }
    __syncthreads();

    const int m  = lane & 15;            // batch row handled by this lane
    const int kh = lane >> 4;            // K half: 0 -> K=0..1, 1 -> K=2..3
    const int stepsPerWave = L >> 4;     // (L/4 positions per wave) / (K=4)
    const int c0 = wave * stepsPerWave;

    v8f accN = {};                       // f32 16x16 accumulators
    v8f accC = {};
    v2f ones; ones[0] = 1.0f; ones[1] = 1.0f;   // B = all-ones (layout-free)

    for (int c = 0; c < stepsPerWave; ++c) {
        const int base = ((c0 + c) << 2) + (kh << 1);
        const float2 a = *(const float2*)(wsN + (size_t)m * L + base);
        const float2 q = *(const float2*)(wsM + (size_t)m * L + base);
        v2f av; av[0] = a.x; av[1] = a.y;
        v2f qv; qv[0] = q.x; qv[1] = q.y;
        // D = A x ones + C  ==> per-row K-sum accumulation, exact f32
        accN = __builtin_amdgcn_wmma_f32_16x16x4_f32(
                   false, av, false, ones, (short)0, accN, false, false);
        accC = __builtin_amdgcn_wmma_f32_16x16x4_f32(
                   false, qv, false, ones, (short)0, accC, false, false);
    }

    // Column n=0 of D lives in lane 0 (M=0..7) and lane 16 (M=8..15).
    if (lane == 0) {
        #pragma unroll
        for (int j = 0; j < 8; ++j) {
            atomicAdd(&sN[j], accN[j]);
            atomicAdd(&sC[j], accC[j]);
        }
    } else if (lane == 16) {
        #pragma unroll
        for (int j = 0; j < 8; ++j) {
            atomicAdd(&sN[8 + j], accN[j]);
            atomicAdd(&sC[8 + j], accC[j]);
        }
    }
    __syncthreads();

    // Per-row finalize: mean over unk positions, or END-token fallback.
    if (tid < 16) {
        const int b   = tid;
        const float cnt = sC[b];
        float sent;
        if (cnt > 0.0f) {
            sent = sN[b] / fmaxf(cnt, 1.0f);
        } else {
            const int fi   = seqlen[b] + 2;
            const float fp = logits[(long long)(b * L + fi) * V + END_ID];
            sent = -logf(fp);
        }
        const float act = (inserted[b] < seqlen[b]) ? 1.0f : 0.0f;
        sLoss[b] = sent * act;
        sAct[b]  = act;
    }
    __syncthreads();

    if (tid == 0) {
        float s1 = 0.0f, s2 = 0.0f;
        #pragma unroll
        for (int b = 0; b < 16; ++b) { s1 += sLoss[b]; s2 += sAct[b]; }
        out[0] = s1 / fmaxf(s2, 1.0f);
    }
}

// ---------------------------------------------------------------------------
extern "C" void kernel_launch(void* const* d_in, const int* in_sizes, int n_in,
                              void* d_out, int out_size, void* d_ws, size_t ws_size,
                              hipStream_t stream) {
    const float* logits   = (const float*)d_in[0];
    const int*   ftrgs    = (const int*)d_in[1];
    const int*   targets  = (const int*)d_in[2];
    const int*   seqlen   = (const int*)d_in[3];
    const int*   inserted = (const int*)d_in[4];

    const int       B = in_sizes[3];                                  // 16
    const int       L = in_sizes[1] / B;                              // 512
    const long long V = (long long)in_sizes[0] / ((long long)B * L);  // 32000

    float* wsN = (float*)d_ws;
    float* wsM = wsN + (size_t)B * L;

    ti_loss_gather<<<B, L, 0, stream>>>(logits, ftrgs, targets, wsN, wsM, L, V);
    ti_loss_reduce_wmma<<<1, 128, 0, stream>>>(wsN, wsM, logits, seqlen, inserted,
                                               (float*)d_out, L, V);
}